// CPCFuturePredictor_32521492365460
// MI455X (gfx1250) — compile-verified
//
#include <hip/hip_runtime.h>
#include <hip/hip_bf16.h>

#define B_      16
#define C_      256
#define T_      2048
#define STEPS_  12
#define NNEG_   15
#define OFFSET_ 16
#define COPIES_ 16

typedef __attribute__((ext_vector_type(2))) float v2f;
typedef __attribute__((ext_vector_type(4))) float v4f;
typedef __attribute__((ext_vector_type(8))) float v8f;

#if defined(__gfx1250__) && __has_builtin(__builtin_amdgcn_global_load_async_to_lds_b128)
#define USE_ASYNC_LDS 1
#else
#define USE_ASYNC_LDS 0
#endif

#if USE_ASYNC_LDS
typedef int vi4 __attribute__((vector_size(16)));           // matches builtin param
typedef __attribute__((address_space(1))) vi4* gv4i_p;      // global int4*
typedef __attribute__((address_space(3))) vi4* lv4i_p;      // LDS int4*
typedef __attribute__((address_space(1))) void* gvoid_p;
typedef __attribute__((address_space(3))) void* lvoid_p;
#endif

__device__ __forceinline__ void wait_async_zero() {
#if USE_ASYNC_LDS
#if __has_builtin(__builtin_amdgcn_s_wait_asynccnt)
  __builtin_amdgcn_s_wait_asynccnt(0);
#else
  asm volatile("s_wait_asynccnt 0x0" ::: "memory");
#endif
#endif
}

// ---------------- z[b,c,t] -> zt[b,t,c] (tiled LDS transpose) ----------------
__global__ void k_transpose_z(const float* __restrict__ z, float* __restrict__ zt) {
  __shared__ float tile[32][33];
  const int t0 = blockIdx.x * 32, c0 = blockIdx.y * 32, b = blockIdx.z;
  const int tx = threadIdx.x, ty = threadIdx.y;
#pragma unroll
  for (int j = 0; j < 32; j += 8)
    tile[ty + j][tx] = z[(size_t)(b * C_ + c0 + ty + j) * T_ + t0 + tx];
  __syncthreads();
#pragma unroll
  for (int j = 0; j < 32; j += 8)
    zt[(size_t)(b * T_ + t0 + ty + j) * C_ + c0 + tx] = tile[tx][ty + j];
}

// ------- weight[c,o,i] -> wt2[i][pair-interleaved (c,o)] ---------------------
// wt2[i][ ((c>>1)*C + o)*2 + (c&1) ] : K-pairs adjacent so a WMMA B-fragment
// is one aligned b64 LDS load, and each 16-row K-chunk is 16KB contiguous.
__global__ void k_transpose_w(const float* __restrict__ w, float* __restrict__ wt2) {
  const int c = blockIdx.x, i = blockIdx.y, o = threadIdx.x;
  wt2[(size_t)i * (C_ * C_) + (((c >> 1) * C_ + o) << 1) + (c & 1)] =
      w[((size_t)c * C_ + o) * STEPS_ + i];
}

// ---------------- zero-fill (labels) ----------------
__global__ void k_zero(float* __restrict__ p, int n) {
  const int idx = blockIdx.x * blockDim.x + threadIdx.x;
  if (idx < n) p[idx] = 0.0f;
}

// ---------------- per-step prediction GEMM: zp[b*T+t, o] = zt row . W[i] -----
// M = B*T = 32768, N = 256, K = 256. Block: 64x256 tile, 8 waves, wave: 16x128.
__global__ __launch_bounds__(256) void k_pred_gemm(const float* __restrict__ zt,
                                                   const float* __restrict__ wt2,
                                                   const float* __restrict__ bias,
                                                   float* __restrict__ zp, int step) {
  __shared__ float lw2[16 * 256];  // 16KB K-chunk, pair-interleaved
  const int tid   = threadIdx.x;
  const int wave  = tid >> 5, lane = tid & 31;
  const int lrow  = lane & 15, khalf = lane >> 4;
  const int r0    = blockIdx.x * 64 + 16 * (wave & 3);  // this wave's 16 M-rows
  const int nbase = 128 * (wave >> 2);                  // this wave's 128 N-cols
  const float* wslice = wt2 + (size_t)step * (C_ * C_);

  v8f acc[8];
#pragma unroll
  for (int nt = 0; nt < 8; ++nt)
#pragma unroll
    for (int e = 0; e < 8; ++e) acc[nt][e] = 0.0f;

  for (int kc = 0; kc < C_; kc += 16) {
    __syncthreads();
    {
      const float* src = wslice + (size_t)kc * C_;  // contiguous 4096 floats
#if USE_ASYNC_LDS
#pragma unroll
      for (int q = 0; q < 4; ++q) {
        const float* gp = src + 4 * (tid + 256 * q);
        float*       lp = lw2 + 4 * (tid + 256 * q);
        __builtin_amdgcn_global_load_async_to_lds_b128(
            (gv4i_p)(gvoid_p)(void*)gp,
            (lv4i_p)(lvoid_p)(void*)lp, 0, 0);
      }
      wait_async_zero();
#else
#pragma unroll
      for (int q = 0; q < 4; ++q)
        *(v4f*)(lw2 + 4 * (tid + 256 * q)) =
            *(const v4f*)(src + 4 * (tid + 256 * q));
#endif
    }
    __syncthreads();
#pragma unroll
    for (int ks = 0; ks < 4; ++ks) {
      const int c4 = kc + 4 * ks;
      // A 16x4 f32 frag: lanes 0-15 M rows, K={0,1}; lanes 16-31 K={2,3}
      v2f a = *(const v2f*)(zt + (size_t)(r0 + lrow) * C_ + c4 + 2 * khalf);
#pragma unroll
      for (int nt = 0; nt < 8; ++nt) {
        const int col = nbase + 16 * nt + lrow;
        // B 4x16 frag: one aligned b64 load of the interleaved K-pair
        v2f bb = *(const v2f*)(lw2 + (((2 * ks + khalf) * C_ + col) << 1));
        acc[nt] = __builtin_amdgcn_wmma_f32_16x16x4_f32(
            false, a, false, bb, (short)0, acc[nt], false, false);
      }
    }
  }
  // C/D layout: VGPR e -> M = e + 8*khalf, N = lrow
#pragma unroll
  for (int nt = 0; nt < 8; ++nt) {
    const int col = nbase + 16 * nt + lrow;
    const float bv = bias[col];
#pragma unroll
    for (int e = 0; e < 8; ++e)
      zp[(size_t)(r0 + e + 8 * khalf) * C_ + col] = acc[nt][e] + bv;
  }
}

// ---------------- per-step scoring: one wave per (t, b) ----------------------
// out[base + (t*B+b)*16 + n] = dot(zp[b*T+t, :], target_n row of zt)
__global__ __launch_bounds__(256) void k_score(const float* __restrict__ zp,
                                               const float* __restrict__ zt,
                                               const int* __restrict__ neg,
                                               float* __restrict__ out,
                                               int step, int valid, long base) {
  const int gw   = (int)((blockIdx.x * blockDim.x + threadIdx.x) >> 5);
  const int lane = threadIdx.x & 31;
  if (gw >= valid * B_) return;           // wave-uniform exit
  const int b = gw & (B_ - 1);
  const int t = gw >> 4;                  // B_ == 16
  const int off = step + OFFSET_;

  const float* zr = zp + (size_t)(b * T_ + t) * C_;
  v4f za = *(const v4f*)(zr + 4 * lane);          // c = 4*lane .. +3
  v4f zb = *(const v4f*)(zr + 128 + 4 * lane);    // c = 128+4*lane .. +3

  float res = 0.0f;
#pragma unroll
  for (int n = 0; n < COPIES_; ++n) {
    int m;
    if (n == 0) m = b * T_ + t + off;                                   // positive
    else        m = neg[(size_t)b * (NNEG_ * T_) + (n - 1) * T_ + t + off];
    const float* tr = zt + (size_t)m * C_;        // contiguous 1KB row
    v4f ta = *(const v4f*)(tr + 4 * lane);
    v4f tb = *(const v4f*)(tr + 128 + 4 * lane);
    float s = za.x * ta.x + za.y * ta.y + za.z * ta.z + za.w * ta.w
            + zb.x * tb.x + zb.y * tb.y + zb.z * tb.z + zb.w * tb.w;
#pragma unroll
    for (int d = 16; d >= 1; d >>= 1) s += __shfl_xor(s, d, 32);
    if (lane == n) res = s;
  }
  if (lane < COPIES_)
    out[base + ((size_t)t * B_ + b) * COPIES_ + lane] = res;
}

extern "C" void kernel_launch(void* const* d_in, const int* in_sizes, int n_in,
                              void* d_out, int out_size, void* d_ws, size_t ws_size,
                              hipStream_t stream) {
  const float* z    = (const float*)d_in[0];
  const float* w    = (const float*)d_in[1];
  const float* bias = (const float*)d_in[2];
  const int*   neg  = (const int*)d_in[3];
  float*       out  = (float*)d_out;

  float* zt  = (float*)d_ws;                  // B*T*C floats   (33.5 MB)
  float* zp  = zt + (size_t)B_ * T_ * C_;     // B*T*C floats   (33.5 MB, reused per step)
  float* wt2 = zp + (size_t)B_ * T_ * C_;     // STEPS*C*C floats (3 MB)

  k_transpose_z<<<dim3(T_ / 32, C_ / 32, B_), dim3(32, 8), 0, stream>>>(z, zt);
  k_transpose_w<<<dim3(C_, STEPS_), C_, 0, stream>>>(w, wt2);

  long pred = 0;
  for (int i = 0; i < STEPS_; ++i) pred += (long)(T_ - (i + OFFSET_)) * B_ * COPIES_;
  const long labels = pred / COPIES_;
  k_zero<<<(int)((labels + 255) / 256), 256, 0, stream>>>(out + pred, (int)labels);

  long base = 0;
  for (int i = 0; i < STEPS_; ++i) {
    k_pred_gemm<<<dim3((B_ * T_) / 64), 256, 0, stream>>>(zt, wt2, bias, zp, i);
    const int valid  = T_ - (i + OFFSET_);
    const int blocks = (valid * B_ + 7) / 8;   // 8 waves / block
    k_score<<<blocks, 256, 0, stream>>>(zp, zt, neg, out, i, valid, base);
    base += (long)valid * B_ * COPIES_;
  }
}